// simpleGCN_APPNP_35656818491450
// MI455X (gfx1250) — compile-verified
//
#include <hip/hip_runtime.h>

#define NUM_GENES 20000
#define NUM_CLASS 32
#define BATCH_SZ  8
#define NN        (NUM_GENES * BATCH_SZ)   /* 160000 nodes */
#define NE        2560000                  /* edges */
#define KSTEPS    10
#define ALPHA     0.1f

typedef float v2f __attribute__((ext_vector_type(2)));
typedef float v8f __attribute__((ext_vector_type(8)));

// ---------------------------------------------------------------------------
// Kernel 1: feature build + 2-layer MLP via V_WMMA_F32_16X16X4_F32.
// One wave per 16-row tile. K=32 done as 8 chained 16x16x4 WMMAs per N-tile.
// ---------------------------------------------------------------------------
__global__ __launch_bounds__(128) void mlp_wmma_kernel(
    const float* __restrict__ x,       // [NN]
    const float* __restrict__ emb,     // [NUM_GENES, 32]
    const float* __restrict__ ebias,   // [NUM_GENES]
    const float* __restrict__ W1,      // [32,32] row-major (used as h@W1^T)
    const float* __restrict__ bias1,   // [32]
    const float* __restrict__ W2,      // [32,32]
    const float* __restrict__ bias2,   // [32]
    float* __restrict__ h0,            // [NN,32] out
    float* __restrict__ hcur)          // [NN,32] out (propagation carry)
{
    __shared__ float lds[4][16 * 33];      // per-wave staging, stride 33

    const int lane = threadIdx.x & 31;
    const int wave = threadIdx.x >> 5;
    const int tile = blockIdx.x * 4 + wave;
    const int base = tile * 16;
    const int l15  = lane & 15;
    const int hi   = lane >> 4;            // 0 or 1
    const int kp   = hi * 2;               // K sub-offset: 0 or 2

    // --- raw features as A-fragments: h[m,k] = emb[gene,k]*x[m] + ebias[gene]
    const int   row  = base + l15;
    const int   gene = row % NUM_GENES;
    const float xv   = x[row];
    const float bv   = ebias[gene];

    v2f a[8];
#pragma unroll
    for (int kk = 0; kk < 8; ++kk) {
        const float* p = emb + (size_t)gene * NUM_CLASS + 4 * kk + kp;
        a[kk].x = p[0] * xv + bv;
        a[kk].y = p[1] * xv + bv;
    }

    // --- GEMM1: relu(h @ W1^T + b1) ---
    v8f acc0 = {};
    v8f acc1 = {};
#pragma unroll
    for (int kk = 0; kk < 8; ++kk) {
        const int kcol = 4 * kk + kp;
        v2f bf0, bf1;
        bf0.x = W1[(l15)      * NUM_CLASS + kcol];
        bf0.y = W1[(l15)      * NUM_CLASS + kcol + 1];
        bf1.x = W1[(16 + l15) * NUM_CLASS + kcol];
        bf1.y = W1[(16 + l15) * NUM_CLASS + kcol + 1];
        acc0 = __builtin_amdgcn_wmma_f32_16x16x4_f32(false, a[kk], false, bf0,
                                                     (short)0, acc0, false, false);
        acc1 = __builtin_amdgcn_wmma_f32_16x16x4_f32(false, a[kk], false, bf1,
                                                     (short)0, acc1, false, false);
    }

    // bias + relu, stage tile to LDS in row-major (stride 33 vs 64 banks)
    float* l = lds[wave];
    const float bb0 = bias1[l15];
    const float bb1 = bias1[16 + l15];
#pragma unroll
    for (int j = 0; j < 8; ++j) {
        const int m  = j + 8 * hi;
        float v0 = acc0[j] + bb0;
        float v1 = acc1[j] + bb1;
        l[m * 33 + l15]      = v0 > 0.f ? v0 : 0.f;
        l[m * 33 + 16 + l15] = v1 > 0.f ? v1 : 0.f;
    }

    // re-gather as A-fragments (same wave owns the region; DS ops are in-order)
#pragma unroll
    for (int kk = 0; kk < 8; ++kk) {
        a[kk].x = l[l15 * 33 + 4 * kk + kp];
        a[kk].y = l[l15 * 33 + 4 * kk + kp + 1];
    }

    // --- GEMM2: h @ W2^T + b2 ---
    v8f c0 = {};
    v8f c1 = {};
#pragma unroll
    for (int kk = 0; kk < 8; ++kk) {
        const int kcol = 4 * kk + kp;
        v2f bf0, bf1;
        bf0.x = W2[(l15)      * NUM_CLASS + kcol];
        bf0.y = W2[(l15)      * NUM_CLASS + kcol + 1];
        bf1.x = W2[(16 + l15) * NUM_CLASS + kcol];
        bf1.y = W2[(16 + l15) * NUM_CLASS + kcol + 1];
        c0 = __builtin_amdgcn_wmma_f32_16x16x4_f32(false, a[kk], false, bf0,
                                                   (short)0, c0, false, false);
        c1 = __builtin_amdgcn_wmma_f32_16x16x4_f32(false, a[kk], false, bf1,
                                                   (short)0, c1, false, false);
    }

    const float cb0 = bias2[l15];
    const float cb1 = bias2[16 + l15];
#pragma unroll
    for (int j = 0; j < 8; ++j) {
        const int m = j + 8 * hi;
        const size_t r = (size_t)(base + m) * NUM_CLASS;
        float v0 = c0[j] + cb0;
        float v1 = c1[j] + cb1;
        h0[r + l15]        = v0;
        hcur[r + l15]      = v0;
        h0[r + 16 + l15]   = v1;
        hcur[r + 16 + l15] = v1;
    }
}

// ---------------------------------------------------------------------------
// Degree / norm precompute
// ---------------------------------------------------------------------------
__global__ void deg_init_kernel(float* __restrict__ deg)
{
    int i = blockIdx.x * blockDim.x + threadIdx.x;
    if (i < NN) deg[i] = 1.0f;                 // self-loop
}

__global__ void deg_count_kernel(const int* __restrict__ col, float* __restrict__ deg)
{
    int e = blockIdx.x * blockDim.x + threadIdx.x;
    if (e < NE) atomicAdd(&deg[col[e]], 1.0f);
}

__global__ void dinv_kernel(float* __restrict__ deg)   // in-place rsqrt
{
    int i = blockIdx.x * blockDim.x + threadIdx.x;
    if (i < NN) deg[i] = rsqrtf(deg[i]);
}

__global__ void norm_kernel(const int* __restrict__ row, const int* __restrict__ col,
                            const float* __restrict__ dinv, float* __restrict__ norm)
{
    int e = blockIdx.x * blockDim.x + threadIdx.x;
    if (e < NE) norm[e] = (1.0f - ALPHA) * dinv[row[e]] * dinv[col[e]];
}

// ---------------------------------------------------------------------------
// Propagation step init: out = ALPHA*h0 + 0.9*dinv^2*h  (self-loop folded,
// no atomics). float4-vectorized: b128 loads/stores over L2-resident state.
// ---------------------------------------------------------------------------
__global__ void selfinit_kernel(const float4* __restrict__ h0,
                                const float4* __restrict__ hin,
                                const float* __restrict__ dinv,
                                float4* __restrict__ hout)
{
    int i = blockIdx.x * blockDim.x + threadIdx.x;   // over NN*8 float4s
    if (i < NN * (NUM_CLASS / 4)) {
        float d = dinv[i >> 3];                      // 8 float4s per node
        float s = (1.0f - ALPHA) * d * d;
        float4 a = h0[i];
        float4 b = hin[i];
        float4 o;
        o.x = ALPHA * a.x + s * b.x;
        o.y = ALPHA * a.y + s * b.y;
        o.z = ALPHA * a.z + s * b.z;
        o.w = ALPHA * a.w + s * b.w;
        hout[i] = o;
    }
}

// One wave per edge, 32 lanes = 32 channels: coalesced 128B gather + 128B
// contiguous atomic region (node state lives in the 192MB L2). Edge index is
// wave-uniform: force it into an SGPR so row/col/norm metadata become scalar
// (s_load) instead of 3 redundant per-lane VMEM loads per edge.
__global__ __launch_bounds__(256) void edge_scatter_kernel(
    const int* __restrict__ row, const int* __restrict__ col,
    const float* __restrict__ norm,
    const float* __restrict__ hin, float* __restrict__ hout)
{
    const int c = threadIdx.x & 31;
    int e = blockIdx.x * 8 + (threadIdx.x >> 5);     // wave-uniform on wave32
    e = __builtin_amdgcn_readfirstlane(e);           // pin to SGPR
    if (e < NE) {
        int   r = row[e];                            // scalar loads (uniform addr)
        int   t = col[e];
        float w = norm[e];                           // includes (1-ALPHA)
        atomicAdd(&hout[(size_t)t * NUM_CLASS + c],
                  w * hin[(size_t)r * NUM_CLASS + c]);
    }
}

// ---------------------------------------------------------------------------
// Mean pool: one block per graph. float4 loads; thread t owns channel group
// (t%8)*4 .. (t%8)*4+3 (invariant since 256 % 8 == 0), fully coalesced b128.
// ---------------------------------------------------------------------------
__global__ __launch_bounds__(256) void pool_kernel(const float4* __restrict__ h,
                                                   float* __restrict__ out)
{
    __shared__ float red[256 * 4];
    const int b = blockIdx.x;
    const int t = threadIdx.x;
    const size_t base = (size_t)b * NUM_GENES * (NUM_CLASS / 4);
    float4 acc = make_float4(0.f, 0.f, 0.f, 0.f);
    for (int i = t; i < NUM_GENES * (NUM_CLASS / 4); i += 256) {
        float4 v = h[base + i];                      // i%8 == t%8 always
        acc.x += v.x; acc.y += v.y; acc.z += v.z; acc.w += v.w;
    }
    red[t * 4 + 0] = acc.x;
    red[t * 4 + 1] = acc.y;
    red[t * 4 + 2] = acc.z;
    red[t * 4 + 3] = acc.w;
    __syncthreads();
    if (t < 32) {
        // channel t: partials live in threads with (tt%8) == t/4, component t%4
        float s = 0.f;
#pragma unroll
        for (int j = 0; j < 32; ++j)
            s += red[((t >> 2) + 8 * j) * 4 + (t & 3)];
        out[b * NUM_CLASS + t] = s / (float)NUM_GENES;
    }
}

// ---------------------------------------------------------------------------
extern "C" void kernel_launch(void* const* d_in, const int* in_sizes, int n_in,
                              void* d_out, int out_size, void* d_ws, size_t ws_size,
                              hipStream_t stream)
{
    const float* x     = (const float*)d_in[0];
    const int*   ei    = (const int*)d_in[1];    // [2, NE]
    /* d_in[2] = batch (layout known: b = node / NUM_GENES) */
    const float* emb   = (const float*)d_in[3];
    const float* ebias = (const float*)d_in[4];
    const float* W1    = (const float*)d_in[5];
    const float* b1    = (const float*)d_in[6];
    const float* W2    = (const float*)d_in[7];
    const float* b2    = (const float*)d_in[8];

    const int* row = ei;
    const int* col = ei + NE;

    float* ws   = (float*)d_ws;
    float* h0   = ws;  ws += (size_t)NN * NUM_CLASS;
    float* hA   = ws;  ws += (size_t)NN * NUM_CLASS;
    float* hB   = ws;  ws += (size_t)NN * NUM_CLASS;
    float* dinv = ws;  ws += NN;                    // used as deg, then rsqrt'd
    float* norm = ws;                               // NE floats

    // MLP (WMMA) -> h0 and initial carry hA
    mlp_wmma_kernel<<<NN / 16 / 4, 128, 0, stream>>>(x, emb, ebias, W1, b1, W2, b2,
                                                     h0, hA);

    // gcn_norm with self-loops
    deg_init_kernel<<<(NN + 255) / 256, 256, 0, stream>>>(dinv);
    deg_count_kernel<<<(NE + 255) / 256, 256, 0, stream>>>(col, dinv);
    dinv_kernel<<<(NN + 255) / 256, 256, 0, stream>>>(dinv);
    norm_kernel<<<(NE + 255) / 256, 256, 0, stream>>>(row, col, dinv, norm);

    // K power-iteration steps, ping-pong
    float* cur = hA;
    float* nxt = hB;
    const int n_vec4 = NN * (NUM_CLASS / 4);
    for (int s = 0; s < KSTEPS; ++s) {
        selfinit_kernel<<<(n_vec4 + 255) / 256, 256, 0, stream>>>(
            (const float4*)h0, (const float4*)cur, dinv, (float4*)nxt);
        edge_scatter_kernel<<<(NE + 7) / 8, 256, 0, stream>>>(row, col, norm, cur, nxt);
        float* tmp = cur; cur = nxt; nxt = tmp;
    }

    pool_kernel<<<BATCH_SZ, 256, 0, stream>>>((const float4*)cur, (float*)d_out);
}